// FusedGCNLayer_8744553415191
// MI455X (gfx1250) — compile-verified
//
#include <hip/hip_runtime.h>
#include <hip/hip_bf16.h>

typedef __attribute__((ext_vector_type(2))) float v2f;
typedef __attribute__((ext_vector_type(8))) float v8f;

#define N_NODES 65536
#define N_EDGES 524288
#define FEAT_DIM 512
#define EMBED_DIM 512

// ---------------------------------------------------------------------------
// Kernel 1: h = x @ W^T using V_WMMA_F32_16X16X4_F32 (native fp32 WMMA).
// One wave -> 16(M) x 64(N) output tile: 4 accumulators, A reused 4x.
// A tile (16x4, MxK):  lane l<16 holds {A[m0+l][k], A[m0+l][k+1]},
//                      lane l>=16 holds {A[m0+l-16][k+2], A[m0+l-16][k+3]}
// B tile (4x16, KxN):  B[k][n] = W[n][k]; same lane pattern over W rows.
// Both are a single contiguous float2 (b64) load per lane.
// C/D (16x16 f32): VGPR r, lanes 0-15 -> (M=r, N=lane); lanes 16-31 -> (M=r+8).
// ---------------------------------------------------------------------------
__global__ __launch_bounds__(256) void gcn_gemm_wmma(
    const float* __restrict__ x, const float* __restrict__ w,
    float* __restrict__ h) {
  const int lane = threadIdx.x & 31;
  const int waveInBlock = threadIdx.x >> 5;
  const int waveId = blockIdx.x * 8 + waveInBlock;

  const int mTile = waveId >> 3;   // 4096 tiles of 16 rows
  const int nChunk = waveId & 7;   // 8 chunks of 64 cols
  const int m0 = mTile * 16;
  const int n0 = nChunk * 64;

  const int l = lane & 15;
  const int half = lane >> 4;

  const float* aPtr = x + (size_t)(m0 + l) * FEAT_DIM + 2 * half;
  const float* bPtr = w + (size_t)(n0 + l) * FEAT_DIM + 2 * half;

  v8f acc0 = {}, acc1 = {}, acc2 = {}, acc3 = {};

#pragma unroll 4
  for (int k = 0; k < FEAT_DIM; k += 4) {
    v2f a  = *(const v2f*)(aPtr + k);
    v2f b0 = *(const v2f*)(bPtr + k);
    v2f b1 = *(const v2f*)(bPtr + 16 * FEAT_DIM + k);
    v2f b2 = *(const v2f*)(bPtr + 32 * FEAT_DIM + k);
    v2f b3 = *(const v2f*)(bPtr + 48 * FEAT_DIM + k);
    acc0 = __builtin_amdgcn_wmma_f32_16x16x4_f32(false, a, false, b0, (short)0, acc0, false, false);
    acc1 = __builtin_amdgcn_wmma_f32_16x16x4_f32(false, a, false, b1, (short)0, acc1, false, false);
    acc2 = __builtin_amdgcn_wmma_f32_16x16x4_f32(false, a, false, b2, (short)0, acc2, false, false);
    acc3 = __builtin_amdgcn_wmma_f32_16x16x4_f32(false, a, false, b3, (short)0, acc3, false, false);
  }

  // Store: VGPR r of acc_j -> h[m0 + r + 8*half][n0 + 16*j + l]
  float* o = h + (size_t)(m0 + 8 * half) * EMBED_DIM + n0 + l;
#pragma unroll
  for (int r = 0; r < 8; ++r) {
    o[(size_t)r * EMBED_DIM +  0] = acc0[r];
    o[(size_t)r * EMBED_DIM + 16] = acc1[r];
    o[(size_t)r * EMBED_DIM + 32] = acc2[r];
    o[(size_t)r * EMBED_DIM + 48] = acc3[r];
  }
}

// ---------------------------------------------------------------------------
// Kernel 2: zero the output (harness poisons d_out before timing).
// ---------------------------------------------------------------------------
__global__ __launch_bounds__(256) void gcn_zero(float4* __restrict__ p, int n4) {
  int i = blockIdx.x * 256 + threadIdx.x;
  if (i < n4) p[i] = make_float4(0.f, 0.f, 0.f, 0.f);
}

// ---------------------------------------------------------------------------
// Hardware fp32 scatter-add: guaranteed global_atomic_add_f32 (no CAS loop).
// Non-returning atomic -> tracked by STOREcnt; implicit s_wait_idle at endpgm.
// ---------------------------------------------------------------------------
__device__ __forceinline__ void hw_atomic_add_f32(float* p, float v) {
  asm volatile("global_atomic_add_f32 %0, %1, off"
               :
               : "v"(p), "v"(v)
               : "memory");
}

// ---------------------------------------------------------------------------
// Kernel 3: edge scatter: out[row[e]] += vals[e] * h[col[e]]
// 2 edges per 256-thread block (128 threads/edge); float4 (b128) gather of the
// L2-resident h row, 4 hardware fp32 atomic adds per thread.
// ---------------------------------------------------------------------------
__global__ __launch_bounds__(256) void gcn_scatter(
    const float* __restrict__ h, const int* __restrict__ row,
    const int* __restrict__ col, const float* __restrict__ vals,
    float* __restrict__ out) {
  const int t   = threadIdx.x;
  const int sub = t >> 7;                  // 0 or 1: which edge in this block
  const int tt  = t & 127;                 // 128 threads x float4 = 512 cols
  const int e   = blockIdx.x * 2 + sub;

  const int r = row[e];
  const int c = col[e];
  const float v = vals[e];

  const float4* __restrict__ src = (const float4*)(h + (size_t)c * EMBED_DIM);
  float* __restrict__ dst = out + (size_t)r * EMBED_DIM + 4 * tt;

  float4 d = src[tt];
  hw_atomic_add_f32(dst + 0, d.x * v);
  hw_atomic_add_f32(dst + 1, d.y * v);
  hw_atomic_add_f32(dst + 2, d.z * v);
  hw_atomic_add_f32(dst + 3, d.w * v);
}

extern "C" void kernel_launch(void* const* d_in, const int* in_sizes, int n_in,
                              void* d_out, int out_size, void* d_ws, size_t ws_size,
                              hipStream_t stream) {
  const float* x    = (const float*)d_in[0];
  const float* w    = (const float*)d_in[1];
  const int*   row  = (const int*)d_in[2];
  const int*   col  = (const int*)d_in[3];
  const float* vals = (const float*)d_in[4];
  float* out = (float*)d_out;
  float* h   = (float*)d_ws;  // needs N_NODES*EMBED_DIM*4 = 128 MB scratch

  // GEMM: 4096 m-tiles * 8 n-chunks = 32768 waves, 8 waves/block.
  gcn_gemm_wmma<<<4096, 256, 0, stream>>>(x, w, h);

  // Zero output: 65536*512 floats = 8388608 float4.
  const int n4 = N_NODES * EMBED_DIM / 4;
  gcn_zero<<<n4 / 256, 256, 0, stream>>>((float4*)out, n4);

  // Scatter-add over edges, 2 edges per block.
  gcn_scatter<<<N_EDGES / 2, 256, 0, stream>>>(h, row, col, vals, out);
}